// SharedResidualQuantizer_38774964748662
// MI455X (gfx1250) — compile-verified
//
#include <hip/hip_runtime.h>
#include <hip/hip_bf16.h>

// ---------------------------------------------------------------------------
// Residual VQ (8 levels, K=8192, d=64, N=32768) for MI455X / gfx1250.
//  - distance GEMM on the matrix engine: V_WMMA_F32_16X16X4_F32 (fp32-exact)
//  - double-buffered codebook staging via GLOBAL_LOAD_ASYNC_TO_LDS_B128
//    (ASYNCcnt), overlapping L2 traffic with WMMA compute
//  - two independent WMMA accumulator chains per loop for XDL ILP
// ---------------------------------------------------------------------------

typedef __attribute__((ext_vector_type(2))) float v2f;
typedef __attribute__((ext_vector_type(8))) float v8f;

#define DEPTH       8
#define NEMBED      8192
#define EDIM        64
#define NROWS       32768           // B*H*W = 32*32*32
#define TILE_CODES  128             // codebook rows staged in LDS per block
#define ROW_PAD     68              // 64 floats + 4 pad (bank-conflict-free b64)
#define NUM_KB      (NEMBED / TILE_CODES)   // 64
#define WG_THREADS  256             // 8 wave32
#define ROWS_PER_WG 128             // 8 waves * 16 rows
#define NUM_WG      (NROWS / ROWS_PER_WG)   // 256

// d_out layout (floats), concatenated reference return order:
#define OUT_VQGAN   0               // quants[0]  (B,C,H,W)  2097152
#define OUT_VQKD    2097152         // quants[7]  (B,C,H,W)  2097152
#define OUT_COMMIT  4194304
#define OUT_KD_DN   4194305
#define OUT_GAN_DN  4194306
#define OUT_CODES   4194307         // (B,h,w,8)             262144

// d_ws layout (floats):
#define WS_CSQ      0               // [8][8192] per-code ||c||^2
#define WS_DNORM    65536           // [256 wg][8 depth] partial sum(dist^2)
#define WS_COMMIT   67584           // [256 wg][8 depth] partial sum(residual^2)

// ---------------------------------------------------------------------------
// 1) precompute ||c||^2 for every code of every codebook
// ---------------------------------------------------------------------------
__global__ __launch_bounds__(256) void csq_kernel(const float* __restrict__ cb,
                                                  float* __restrict__ csq) {
    int i = blockIdx.x * 256 + threadIdx.x;        // 0 .. 65535
    const float4* row = (const float4*)(cb + (size_t)i * EDIM);
    float s = 0.f;
#pragma unroll
    for (int q = 0; q < 16; ++q) {
        float4 v = row[q];
        s += v.x * v.x + v.y * v.y + v.z * v.z + v.w * v.w;
    }
    csq[i] = s;
}

// ---------------------------------------------------------------------------
// 2) main residual-quantization kernel
// ---------------------------------------------------------------------------
__global__ __launch_bounds__(WG_THREADS) void rvq_kernel(
        const float* __restrict__ x,          // (32, 64, 32, 32)
        const float* __restrict__ codebooks,  // (8, 8192, 64)
        float* __restrict__ out,
        float* __restrict__ ws) {

    __shared__ __align__(16) float tile[2][TILE_CODES * ROW_PAD]; // dbl-buffered slab
    __shared__ float rsq_s[8][16];                 // per-wave row ||r||^2
    __shared__ int   codes_s[8][16];               // per-wave argmin codes
    __shared__ float wredN[8];                     // per-wave dist^2 sums
    __shared__ float wredC[8];                     // per-wave residual^2 sums

    const int tid  = threadIdx.x;
    const int lane = tid & 31;
    const int wave = tid >> 5;
    const int hi   = lane >> 4;      // 0: K sub 0/1, rows 0-7 ; 1: K sub 2/3, rows 8-15
    const int M    = lane & 15;      // row within wave (A/B lane row/col)
    const int wg   = blockIdx.x;
    const int rowBase = wg * ROWS_PER_WG + wave * 16;
    const int n    = rowBase + M;    // this lane's global row (flat b*h*w)
    const int b    = n >> 10;
    const int hw   = n & 1023;

    const float* csq = ws + WS_CSQ;

    // async stage of one 128x64 slab into an LDS buffer (8 b128 ops / thread)
    auto issue_stage = [&](float* buf, const float* slab) {
#pragma unroll
        for (int it = 0; it < 8; ++it) {
            int i  = tid + it * 256;              // 2048 float4 chunks
            int r  = i >> 4;
            int c4 = i & 15;
            unsigned     ldso = (unsigned)(size_t)(&buf[r * ROW_PAD + c4 * 4]);
            const float* g    = slab + r * EDIM + c4 * 4;
            asm volatile("global_load_async_to_lds_b128 %0, %1, off"
                         :: "v"(ldso), "v"(g) : "memory");
        }
    };

    // A-fragment register images of residual / cumulative quant.
    // Lane owns k = 4*c + 2*hi + j  for c in [0,16), j in {0,1}.
    float resA[32];
    float aggA[32];
#pragma unroll
    for (int c = 0; c < 16; ++c) {
#pragma unroll
        for (int j = 0; j < 2; ++j) {
            int k = 4 * c + 2 * hi + j;
            resA[2 * c + j] = x[((size_t)(b * 64 + k) << 10) + hw];
            aggA[2 * c + j] = 0.f;
        }
    }

    for (int t = 0; t < DEPTH; ++t) {
        const float* cb_t = codebooks + (size_t)t * NEMBED * EDIM;

        // ---- per-row ||residual||^2 (needed for distances) ----
        float part = 0.f;
#pragma unroll
        for (int i = 0; i < 32; ++i) part = fmaf(resA[i], resA[i], part);
        float full = part + __shfl_xor(part, 16, 32);
        if (lane < 16) rsq_s[wave][lane] = full;
        __syncthreads();
        float rsq8[8];
#pragma unroll
        for (int j = 0; j < 8; ++j) rsq8[j] = rsq_s[wave][j + 8 * hi];

        float best[8];
        int   bidx[8];
#pragma unroll
        for (int j = 0; j < 8; ++j) { best[j] = 3.4e38f; bidx[j] = 0; }
        float dsum = 0.f;

        // prologue: stage first slab
        issue_stage(tile[0], cb_t);

        for (int kb = 0; kb < NUM_KB; ++kb) {
            const int cur = kb & 1;
            // our outstanding async loads target tile[cur]; wait + make visible
            asm volatile("s_wait_asynccnt 0x0" ::: "memory");
            __syncthreads();
            // overlap: stage next slab while computing on this one
            if (kb + 1 < NUM_KB)
                issue_stage(tile[1 - cur],
                            cb_t + (size_t)(kb + 1) * TILE_CODES * EDIM);

            const float* tbuf = tile[cur];
            // ---- 4 pairs of 16-code sub-tiles: two independent WMMA chains ----
            for (int sub = 0; sub < 8; sub += 2) {
                v8f acc0 = {};
                v8f acc1 = {};
                const float* brow0 = &tbuf[(sub * 16 + M) * ROW_PAD + 2 * hi];
                const float* brow1 = &tbuf[((sub + 1) * 16 + M) * ROW_PAD + 2 * hi];
#pragma unroll
                for (int c = 0; c < 16; ++c) {
                    v2f a;  a.x  = resA[2 * c];    a.y  = resA[2 * c + 1];
                    v2f b0; b0.x = brow0[4 * c];   b0.y = brow0[4 * c + 1];
                    v2f b1; b1.x = brow1[4 * c];   b1.y = brow1[4 * c + 1];
                    acc0 = __builtin_amdgcn_wmma_f32_16x16x4_f32(
                            false, a, false, b0, (short)0, acc0, false, false);
                    acc1 = __builtin_amdgcn_wmma_f32_16x16x4_f32(
                            false, a, false, b1, (short)0, acc1, false, false);
                }
                int   N0  = kb * TILE_CODES + sub * 16 + M;
                int   N1  = N0 + 16;
                float cs0 = csq[t * NEMBED + N0];
                float cs1 = csq[t * NEMBED + N1];
#pragma unroll
                for (int j = 0; j < 8; ++j) {
                    float d0 = fmaf(-2.f, acc0[j], rsq8[j] + cs0);
                    float d1 = fmaf(-2.f, acc1[j], rsq8[j] + cs1);
                    dsum = fmaf(d0, d0, dsum);
                    dsum = fmaf(d1, d1, dsum);
                    if (d0 < best[j]) { best[j] = d0; bidx[j] = N0; }
                    if (d1 < best[j]) { best[j] = d1; bidx[j] = N1; }
                }
            }
        }

        // ---- argmin across the 16 lanes of each half (first-index ties) ----
#pragma unroll
        for (int off = 1; off < 16; off <<= 1) {
#pragma unroll
            for (int j = 0; j < 8; ++j) {
                float od = __shfl_xor(best[j], off, 32);
                int   oi = __shfl_xor(bidx[j], off, 32);
                if (od < best[j] || (od == best[j] && oi < bidx[j])) {
                    best[j] = od; bidx[j] = oi;
                }
            }
        }
        if (M == 0) {                       // lanes 0 and 16
#pragma unroll
            for (int j = 0; j < 8; ++j) codes_s[wave][8 * hi + j] = bidx[j];
        }
        // ---- wave-reduce sum(dist^2) ----
#pragma unroll
        for (int off = 1; off < 32; off <<= 1) dsum += __shfl_xor(dsum, off, 32);
        if (lane == 0) wredN[wave] = dsum;
        __syncthreads();

        // ---- gather winning code, update residual / agg, commit term ----
        int code = codes_s[wave][M];
        const float* qrow = cb_t + (size_t)code * EDIM;
        float csum = 0.f;
#pragma unroll
        for (int c = 0; c < 16; ++c) {
            int   k  = 4 * c + 2 * hi;
            float q0 = qrow[k];
            float q1 = qrow[k + 1];
            resA[2 * c]     -= q0;  resA[2 * c + 1] -= q1;
            aggA[2 * c]     += q0;  aggA[2 * c + 1] += q1;
            csum = fmaf(resA[2 * c], resA[2 * c], csum);
            csum = fmaf(resA[2 * c + 1], resA[2 * c + 1], csum);
        }
#pragma unroll
        for (int off = 1; off < 32; off <<= 1) csum += __shfl_xor(csum, off, 32);
        if (lane == 0) wredC[wave] = csum;

        // ---- codes output (B,h,w,DEPTH) as float ----
        if (lane < 16)
            out[OUT_CODES + (size_t)(rowBase + lane) * DEPTH + t] =
                (float)codes_s[wave][lane];

        // ---- snapshot outputs: quants[0] and quants[7] in (B,C,H,W) ----
        if (t == 0 || t == DEPTH - 1) {
            float* snap = out + (t == 0 ? OUT_VQGAN : OUT_VQKD);
#pragma unroll
            for (int c = 0; c < 16; ++c) {
#pragma unroll
                for (int j = 0; j < 2; ++j) {
                    int k = 4 * c + 2 * hi + j;
                    snap[((size_t)(b * 64 + k) << 10) + hw] = aggA[2 * c + j];
                }
            }
        }
        __syncthreads();
        if (tid == 0) {                      // deterministic per-WG partials
            float sn = 0.f, sc = 0.f;
#pragma unroll
            for (int w = 0; w < 8; ++w) { sn += wredN[w]; sc += wredC[w]; }
            ws[WS_DNORM  + wg * DEPTH + t] = sn;
            ws[WS_COMMIT + wg * DEPTH + t] = sc;
        }
        __syncthreads();
    }
}

// ---------------------------------------------------------------------------
// 3) deterministic finalize: scalars from per-WG partials
// ---------------------------------------------------------------------------
__global__ void finalize_kernel(const float* __restrict__ ws,
                                float* __restrict__ out) {
    if (blockIdx.x == 0 && threadIdx.x == 0) {
        const float* dn = ws + WS_DNORM;
        const float* cm = ws + WS_COMMIT;
        float dn0 = 0.f, dn7 = 0.f, commit = 0.f;
        for (int wg = 0; wg < NUM_WG; ++wg) {
            dn0 += dn[wg * DEPTH + 0];
            dn7 += dn[wg * DEPTH + DEPTH - 1];
            for (int t = 0; t < DEPTH; ++t) commit += cm[wg * DEPTH + t];
        }
        out[OUT_COMMIT] = commit / (float)((size_t)DEPTH * NROWS * EDIM);
        out[OUT_KD_DN]  = dn7 / (float)NROWS;
        out[OUT_GAN_DN] = dn0 / (float)NROWS;
    }
}

extern "C" void kernel_launch(void* const* d_in, const int* in_sizes, int n_in,
                              void* d_out, int out_size, void* d_ws, size_t ws_size,
                              hipStream_t stream) {
    const float* x  = (const float*)d_in[0];          // (32, 64, 32, 32)
    const float* cb = (const float*)d_in[1];          // (8, 8192, 64)
    float* out = (float*)d_out;
    float* ws  = (float*)d_ws;

    csq_kernel<<<(DEPTH * NEMBED) / 256, 256, 0, stream>>>(cb, ws + WS_CSQ);
    rvq_kernel<<<NUM_WG, WG_THREADS, 0, stream>>>(x, cb, out, ws);
    finalize_kernel<<<1, 32, 0, stream>>>(ws, out);
}